// topk_71846212927558
// MI455X (gfx1250) — compile-verified
//
#include <hip/hip_runtime.h>
#include <math.h>

typedef __attribute__((ext_vector_type(2))) float v2f;
typedef __attribute__((ext_vector_type(8))) float v8f;

// ---------------- constants ----------------
#define Bg        512
#define Nn        1024
#define NUM_NODES (Bg * Nn)      // 524288
#define NE        4194304
#define IN_CH     64
#define EMB       16
#define OUT_CH    10
#define K1        512
#define K2        256

// d_out layout (floats)
#define OUT_OFS_OUT    0
#define OUT_OFS_GEMB   (Bg * OUT_CH)                    // 5120
#define OUT_OFS_PW     (OUT_OFS_GEMB + Bg * EMB)        // 13312
#define OUT_OFS_SCORE1 (OUT_OFS_PW + EMB)               // 13328

// workspace layout (float offsets)
#define WS_YREL1   0u
#define WS_YROOT1  8388608u
#define WS_AGG1    16777216u     // also H1 (in-place)
#define WS_SCORE1  25165824u
#define WS_NODEMAP 25690112u     // int region
#define WS_HP1     26214400u
#define WS_X1      30408704u
// reuse region (valid after conv1-finish consumed yrel1/yroot1)
#define WS_YREL2   0u
#define WS_YROOT2  4194304u
#define WS_AGG2    8388608u      // also H2 (in-place)
#define WS_SCORE2  12582912u
#define WS_X2      12845056u

// ---------------- utility fills ----------------
__global__ void fill_f32(float* __restrict__ p, float v, int n) {
  int i = blockIdx.x * blockDim.x + threadIdx.x;
  if (i < n) p[i] = v;
}
__global__ void fill_i32(int* __restrict__ p, int v, int n) {
  int i = blockIdx.x * blockDim.x + threadIdx.x;
  if (i < n) p[i] = v;
}

// ---------------- WMMA projection: Y = X @ W (two weight mats fused) -------
// X: [nrows, K] f32, W: [K, 16] f32, Y: [nrows, 16] f32.
// One wave per 16-row tile; V_WMMA_F32_16X16X4_F32, K/4 steps, dual accum.
template <int K>
__global__ __launch_bounds__(256) void proj_wmma(
    const float* __restrict__ X, const float* __restrict__ Wr,
    const float* __restrict__ Wo, float* __restrict__ Yr,
    float* __restrict__ Yo, int ntiles) {
  int wave = threadIdx.x >> 5;
  int lane = threadIdx.x & 31;
  int tile = blockIdx.x * 8 + wave;
  if (tile >= ntiles) return;          // uniform per wave: EXEC stays all-ones
  int row0 = tile * 16;
  int m    = lane & 15;                // M for A rows, N column for B/C/D
  int half = lane >> 4;

  v8f accR = {0.f, 0.f, 0.f, 0.f, 0.f, 0.f, 0.f, 0.f};
  v8f accO = {0.f, 0.f, 0.f, 0.f, 0.f, 0.f, 0.f, 0.f};

#pragma unroll
  for (int kk = 0; kk < K; kk += 4) {
    int k0 = kk + half * 2;
    const float* xp = X + (size_t)(row0 + m) * K + k0;
    v2f a;  a.x  = xp[0];              a.y  = xp[1];
    v2f br; br.x = Wr[k0 * 16 + m];    br.y = Wr[(k0 + 1) * 16 + m];
    v2f bo; bo.x = Wo[k0 * 16 + m];    bo.y = Wo[(k0 + 1) * 16 + m];
    accR = __builtin_amdgcn_wmma_f32_16x16x4_f32(false, a, false, br,
                                                 (short)0, accR, false, false);
    accO = __builtin_amdgcn_wmma_f32_16x16x4_f32(false, a, false, bo,
                                                 (short)0, accO, false, false);
  }
#pragma unroll
  for (int v = 0; v < 8; ++v) {
    int row = row0 + v + half * 8;     // C/D layout: vgpr v -> M=v / v+8
    Yr[(size_t)row * 16 + m] = accR[v];
    Yo[(size_t)row * 16 + m] = accO[v];
  }
}

// ---------------- edge aggregation (conv1) ----------------
__global__ void agg_edges1(const int* __restrict__ src,
                           const int* __restrict__ dst,
                           const float* __restrict__ Y,
                           float* __restrict__ AGG, int nE) {
  int e = blockIdx.x * blockDim.x + threadIdx.x;
  if (e >= nE) return;
  const float* ys = Y + (size_t)src[e] * 16;
  float* ad       = AGG + (size_t)dst[e] * 16;
#pragma unroll
  for (int c = 0; c < 16; ++c) atomicAdd(ad + c, ys[c]);
}

// ---------------- edge aggregation (conv2, filtered via node_map) ----------
__global__ void agg_edges2(const int* __restrict__ src,
                           const int* __restrict__ dst,
                           const int* __restrict__ node_map,
                           const float* __restrict__ Y,
                           float* __restrict__ AGG, int nE) {
  int e = blockIdx.x * blockDim.x + threadIdx.x;
  if (e >= nE) return;
  int ns = node_map[src[e]];
  int nd = node_map[dst[e]];
  if (ns < 0 || nd < 0) return;
  const float* ys = Y + (size_t)ns * 16;
  float* ad       = AGG + (size_t)nd * 16;
#pragma unroll
  for (int c = 0; c < 16; ++c) atomicAdd(ad + c, ys[c]);
}

// ---------------- fused conv-finish (relu) + pool score --------------------
__global__ void conv_finish_score(float* __restrict__ AGG,        // in agg / out h
                                  const float* __restrict__ YROOT,
                                  const float* __restrict__ bias,
                                  const float* __restrict__ pw,
                                  float* __restrict__ score, int n) {
  int i = blockIdx.x * blockDim.x + threadIdx.x;
  if (i >= n) return;
  float w[16], wn = 0.f;
#pragma unroll
  for (int c = 0; c < 16; ++c) { w[c] = pw[c]; wn += w[c] * w[c]; }
  float inv = rsqrtf(wn);
  size_t base = (size_t)i * 16;
  float dot = 0.f;
#pragma unroll
  for (int c = 0; c < 16; ++c) {
    float h = AGG[base + c] + YROOT[base + c] + bias[c];
    h = fmaxf(h, 0.f);
    AGG[base + c] = h;
    dot += h * w[c];
  }
  float z = dot * inv;
  score[i] = 1.f / (1.f + expf(-z));
}

// ---------------- top-k pool 1: 1024 -> 512, per-graph block ---------------
__global__ __launch_bounds__(1024) void topk_pool1_kernel(
    const float* __restrict__ score_raw,   // [B*N]
    const float* __restrict__ H,           // [B*N,16]
    float* __restrict__ score_out,         // [B*K1] (d_out region)
    int* __restrict__ node_map,            // [B*N], prefilled -1
    float* __restrict__ HP,                // [B*K1,16]
    float* __restrict__ X1) {              // [B,32]
  __shared__ float s_val[Nn];
  __shared__ int   s_idx[Nn];
  __shared__ float s_hp[K1 * 16];
  __shared__ float s_red[Nn];
  int g = blockIdx.x, t = threadIdx.x;
  s_val[t] = score_raw[g * Nn + t];
  s_idx[t] = t;
  __syncthreads();
  // bitonic sort, descending by value, ties -> lower original index first
  for (int k = 2; k <= Nn; k <<= 1) {
    for (int j = k >> 1; j > 0; j >>= 1) {
      int ixj = t ^ j;
      if (ixj > t) {
        bool desc = ((t & k) == 0);
        float va = s_val[t], vb = s_val[ixj];
        int   ia = s_idx[t], ib = s_idx[ixj];
        bool aFirst = (va > vb) || (va == vb && ia < ib);
        if (desc ? !aFirst : aFirst) {
          s_val[t] = vb; s_val[ixj] = va;
          s_idx[t] = ib; s_idx[ixj] = ia;
        }
      }
      __syncthreads();
    }
  }
  if (t < K1) {
    float v = s_val[t];
    int  li = s_idx[t];
    score_out[g * K1 + t] = v;
    node_map[g * Nn + li] = g * K1 + t;
  }
  __syncthreads();
  // gather + scale kept rows
  for (int eoff = t; eoff < K1 * 16; eoff += Nn) {
    int r = eoff >> 4, c = eoff & 15;
    float hv = H[((size_t)g * Nn + s_idx[r]) * 16 + c] * s_val[r];
    s_hp[eoff] = hv;
    HP[((size_t)g * K1 + r) * 16 + c] = hv;
  }
  __syncthreads();
  // readout: per-channel max + mean over K1 rows
  int c = t & 15, grp = t >> 4;            // 64 groups x 8 rows each
  float mx = -INFINITY, sm = 0.f;
#pragma unroll
  for (int q = 0; q < 8; ++q) {
    float v = s_hp[(grp * 8 + q) * 16 + c];
    mx = fmaxf(mx, v); sm += v;
  }
  s_red[t] = mx; s_val[t] = sm;
  __syncthreads();
  for (int g2 = 32; g2 >= 1; g2 >>= 1) {
    if (grp < g2) {
      s_red[grp * 16 + c] = fmaxf(s_red[grp * 16 + c], s_red[(grp + g2) * 16 + c]);
      s_val[grp * 16 + c] += s_val[(grp + g2) * 16 + c];
    }
    __syncthreads();
  }
  if (t < 16) {
    X1[g * 32 + t]      = s_red[t];
    X1[g * 32 + 16 + t] = s_val[t] * (1.0f / K1);
  }
}

// ---------------- top-k pool 2: 512 -> 256, readout only -------------------
__global__ __launch_bounds__(512) void topk_pool2_kernel(
    const float* __restrict__ score_raw,   // [B*K1]
    const float* __restrict__ H2,          // [B*K1,16]
    float* __restrict__ X2) {              // [B,32]
  __shared__ float s_val[K1];
  __shared__ int   s_idx[K1];
  __shared__ float s_hp[K2 * 16];
  __shared__ float s_red[K1];
  int g = blockIdx.x, t = threadIdx.x;
  s_val[t] = score_raw[g * K1 + t];
  s_idx[t] = t;
  __syncthreads();
  for (int k = 2; k <= K1; k <<= 1) {
    for (int j = k >> 1; j > 0; j >>= 1) {
      int ixj = t ^ j;
      if (ixj > t) {
        bool desc = ((t & k) == 0);
        float va = s_val[t], vb = s_val[ixj];
        int   ia = s_idx[t], ib = s_idx[ixj];
        bool aFirst = (va > vb) || (va == vb && ia < ib);
        if (desc ? !aFirst : aFirst) {
          s_val[t] = vb; s_val[ixj] = va;
          s_idx[t] = ib; s_idx[ixj] = ia;
        }
      }
      __syncthreads();
    }
  }
  for (int eoff = t; eoff < K2 * 16; eoff += K1) {
    int r = eoff >> 4, c = eoff & 15;
    s_hp[eoff] = H2[((size_t)g * K1 + s_idx[r]) * 16 + c] * s_val[r];
  }
  __syncthreads();
  int c = t & 15, grp = t >> 4;            // 32 groups x 8 rows each
  float mx = -INFINITY, sm = 0.f;
#pragma unroll
  for (int q = 0; q < 8; ++q) {
    float v = s_hp[(grp * 8 + q) * 16 + c];
    mx = fmaxf(mx, v); sm += v;
  }
  s_red[t] = mx; s_val[t] = sm;
  __syncthreads();
  for (int g2 = 16; g2 >= 1; g2 >>= 1) {
    if (grp < g2) {
      s_red[grp * 16 + c] = fmaxf(s_red[grp * 16 + c], s_red[(grp + g2) * 16 + c]);
      s_val[grp * 16 + c] += s_val[(grp + g2) * 16 + c];
    }
    __syncthreads();
  }
  if (t < 16) {
    X2[g * 32 + t]      = s_red[t];
    X2[g * 32 + 16 + t] = s_val[t] * (1.0f / K2);
  }
}

// ---------------- per-graph head: lin1 + BN + lin2 + log_softmax -----------
__global__ void head_kernel(const float* __restrict__ X1,
                            const float* __restrict__ X2,
                            const float* __restrict__ l1W, const float* __restrict__ l1b,
                            const float* __restrict__ gmm, const float* __restrict__ bta,
                            const float* __restrict__ mean, const float* __restrict__ var,
                            const float* __restrict__ l2W, const float* __restrict__ l2b,
                            const float* __restrict__ pw, float* __restrict__ out) {
  int g = blockIdx.x * blockDim.x + threadIdx.x;
  if (g < EMB) out[OUT_OFS_PW + g] = pw[g];          // pool1_w passthrough
  if (g >= Bg) return;
  float v[32];
#pragma unroll
  for (int k = 0; k < 32; ++k) v[k] = X1[g * 32 + k] + X2[g * 32 + k];
  float ge[EMB];
#pragma unroll
  for (int j = 0; j < EMB; ++j) {
    float a = l1b[j];
    for (int k = 0; k < 32; ++k) a += v[k] * l1W[k * EMB + j];
    float pre = fmaxf(a, 0.f);
    ge[j] = gmm[j] * (pre - mean[j]) * rsqrtf(var[j] + 1e-5f) + bta[j];
    out[OUT_OFS_GEMB + g * EMB + j] = ge[j];
  }
  float lg[OUT_CH], m = -INFINITY;
#pragma unroll
  for (int o = 0; o < OUT_CH; ++o) {
    float a = l2b[o];
    for (int j = 0; j < EMB; ++j) a += ge[j] * l2W[j * OUT_CH + o];
    lg[o] = a;
    m = fmaxf(m, a);
  }
  float s = 0.f;
#pragma unroll
  for (int o = 0; o < OUT_CH; ++o) s += expf(lg[o] - m);
  float lse = logf(s);
#pragma unroll
  for (int o = 0; o < OUT_CH; ++o) out[OUT_OFS_OUT + g * OUT_CH + o] = lg[o] - m - lse;
}

// ---------------- host launcher ----------------
extern "C" void kernel_launch(void* const* d_in, const int* in_sizes, int n_in,
                              void* d_out, int out_size, void* d_ws, size_t ws_size,
                              hipStream_t stream) {
  const float* x      = (const float*)d_in[0];
  const int*   edges  = (const int*)d_in[1];
  const int*   src    = edges;
  const int*   dst    = edges + NE;
  const float* c1Wrel = (const float*)d_in[2];
  const float* c1Wroot= (const float*)d_in[3];
  const float* c1b    = (const float*)d_in[4];
  const float* p1w    = (const float*)d_in[5];
  const float* c2Wrel = (const float*)d_in[6];
  const float* c2Wroot= (const float*)d_in[7];
  const float* c2b    = (const float*)d_in[8];
  const float* p2w    = (const float*)d_in[9];
  const float* l1W    = (const float*)d_in[10];
  const float* l1b    = (const float*)d_in[11];
  const float* bng    = (const float*)d_in[12];
  const float* bnb    = (const float*)d_in[13];
  const float* bnm    = (const float*)d_in[14];
  const float* bnv    = (const float*)d_in[15];
  const float* l2W    = (const float*)d_in[16];
  const float* l2b    = (const float*)d_in[17];

  float* ws   = (float*)d_ws;
  float* out  = (float*)d_out;

  float* yrel1  = ws + WS_YREL1;
  float* yroot1 = ws + WS_YROOT1;
  float* agg1   = ws + WS_AGG1;        // becomes H1
  float* sc1    = ws + WS_SCORE1;
  int*   nmap   = (int*)(ws + WS_NODEMAP);
  float* hp1    = ws + WS_HP1;
  float* x1     = ws + WS_X1;
  float* yrel2  = ws + WS_YREL2;
  float* yroot2 = ws + WS_YROOT2;
  float* agg2   = ws + WS_AGG2;        // becomes H2
  float* sc2    = ws + WS_SCORE2;
  float* x2     = ws + WS_X2;

  // conv1: project (WMMA), zero agg, scatter edges, finish + scores
  proj_wmma<IN_CH><<<NUM_NODES / 16 / 8, 256, 0, stream>>>(
      x, c1Wrel, c1Wroot, yrel1, yroot1, NUM_NODES / 16);
  fill_f32<<<(NUM_NODES * 16) / 256, 256, 0, stream>>>(agg1, 0.f, NUM_NODES * 16);
  agg_edges1<<<NE / 256, 256, 0, stream>>>(src, dst, yrel1, agg1, NE);
  conv_finish_score<<<NUM_NODES / 256, 256, 0, stream>>>(
      agg1, yroot1, c1b, p1w, sc1, NUM_NODES);

  // pool1 (writes score1 to d_out, builds node_map, pooled features, readout)
  fill_i32<<<NUM_NODES / 256, 256, 0, stream>>>(nmap, -1, NUM_NODES);
  topk_pool1_kernel<<<Bg, Nn, 0, stream>>>(sc1, agg1, out + OUT_OFS_SCORE1,
                                           nmap, hp1, x1);

  // conv2 on pooled graph
  proj_wmma<EMB><<<(Bg * K1) / 16 / 8, 256, 0, stream>>>(
      hp1, c2Wrel, c2Wroot, yrel2, yroot2, (Bg * K1) / 16);
  fill_f32<<<(Bg * K1 * 16) / 256, 256, 0, stream>>>(agg2, 0.f, Bg * K1 * 16);
  agg_edges2<<<NE / 256, 256, 0, stream>>>(src, dst, nmap, yrel2, agg2, NE);
  conv_finish_score<<<(Bg * K1) / 256, 256, 0, stream>>>(
      agg2, yroot2, c2b, p2w, sc2, Bg * K1);

  // pool2 (readout only)
  topk_pool2_kernel<<<Bg, K1, 0, stream>>>(sc2, agg2, x2);

  // head: lin1 + BN + lin2 + log_softmax, plus pool1_w copy
  head_kernel<<<2, 256, 0, stream>>>(x1, x2, l1W, l1b, bng, bnb, bnm, bnv,
                                     l2W, l2b, p1w, out);
}